// MRP_29721173688619
// MI455X (gfx1250) — compile-verified
//
#include <hip/hip_runtime.h>
#include <math.h>

// Problem sizes (fixed by the reference)
constexpr int kB = 8;
constexpr int kN = 8192;
constexpr int kD = 256;
constexpr int kK = 64;
constexpr float kLOG2PI = 1.8378770664093453f;

// Workspace layout (floats).
// GP: emission weights in WMMA-paired layout, both passes contiguous:
//   pass0 = mu*inv_var, pass1 = -0.5*inv_var, each [kD/2][kK][2]:
//   GP[pass*kD*kK + (d>>1)*(2*kK) + col*2 + (d&1)]
constexpr size_t OFF_GP   = 0;
constexpr size_t GP_SIZE  = (size_t)2 * kD * kK;          // 32768 floats = 128 KB
constexpr size_t OFF_C0   = OFF_GP + GP_SIZE;             // [kK]
constexpr size_t OFF_AM1  = OFF_C0 + kK;                  // [kK]
constexpr size_t OFF_BNEG = OFF_AM1 + kK;                 // [kK]
constexpr size_t OFF_PP   = OFF_BNEG + kK;                // [kK][kK] softmax(A), scan-paired:
                                                          //   PP[i*kK + (j&31)*2 + (j>>5)]
constexpr size_t OFF_PI0  = OFF_PP + (size_t)kK * kK;     // [kK]
constexpr size_t OFF_LOGZ = OFF_PI0 + kK;                 // [kB]
constexpr size_t OFF_LL   = OFF_LOGZ + kB;                // [kB][kN][kK]

typedef float v2f __attribute__((ext_vector_type(2)));
typedef float v8f __attribute__((ext_vector_type(8)));
typedef unsigned int v4u __attribute__((ext_vector_type(4)));
typedef int v4i __attribute__((ext_vector_type(4)));
typedef int v8i __attribute__((ext_vector_type(8)));

#if defined(__has_builtin)
#if __has_builtin(__builtin_amdgcn_wmma_f32_16x16x4_f32)
#define HAVE_WMMA_F32X4 1
#endif
#if __has_builtin(__builtin_amdgcn_tensor_load_to_lds) && \
    __has_builtin(__builtin_amdgcn_s_wait_tensorcnt)
#define HAVE_TDM 1
#endif
#endif

// ---------------------------------------------------------------------------
// Kernel 1: per-state precompute (64 threads, one per state k)
// ---------------------------------------------------------------------------
__global__ __launch_bounds__(kK) void prep_kernel(
    const float* __restrict__ A, const float* __restrict__ log_ab,
    const float* __restrict__ mu, const float* __restrict__ log_sigma,
    const float* __restrict__ pi, float* __restrict__ ws) {
  const int k = threadIdx.x;
  float* Wp  = ws + OFF_GP;                    // pass 0
  float* Hp  = ws + OFF_GP + (size_t)kD * kK;  // pass 1
  float* c0  = ws + OFF_C0;
  float* am1 = ws + OFF_AM1;
  float* bng = ws + OFF_BNEG;
  float* PP  = ws + OFF_PP;
  float* pi0 = ws + OFF_PI0;

  float csum = 0.0f;
  for (int d = 0; d < kD; ++d) {
    const float ls = log_sigma[k * kD + d];
    const float iv = expf(-2.0f * ls);
    const float m  = mu[k * kD + d];
    const int   pr = (d >> 1) * (2 * kK) + k * 2 + (d & 1);
    Wp[pr] = m * iv;
    Hp[pr] = -0.5f * iv;
    csum += 0.5f * m * m * iv + ls;
  }
  const float a  = expf(log_ab[k * 2 + 0]);
  const float bb = expf(log_ab[k * 2 + 1]);
  const float nrm = a * logf(bb) - lgammaf(a);
  c0[k]  = nrm - (csum + 0.5f * (float)kD * kLOG2PI);
  am1[k] = a - 1.0f;
  bng[k] = -bb;

  // Row-softmax of A, stored pair-packed for the single-wave scan.
  float mx = -INFINITY;
  for (int j = 0; j < kK; ++j) mx = fmaxf(mx, A[k * kK + j]);
  float s = 0.0f;
  for (int j = 0; j < kK; ++j) s += expf(A[k * kK + j] - mx);
  const float inv = 1.0f / s;
  for (int j = 0; j < kK; ++j)
    PP[k * kK + (j & 31) * 2 + (j >> 5)] = expf(A[k * kK + j] - mx) * inv;

  // pi0 = log_softmax(pi)
  float mp = -INFINITY;
  for (int j = 0; j < kK; ++j) mp = fmaxf(mp, pi[j]);
  float sp = 0.0f;
  for (int j = 0; j < kK; ++j) sp += expf(pi[j] - mp);
  pi0[k] = pi[k] - mp - logf(sp);
}

// ---------------------------------------------------------------------------
// Kernel 2: emission + gamma log-lik. WMMA f32 16x16x4, weights staged in LDS
// by the Tensor Data Mover. Grid: 512 blocks x 256 threads (8 waves); each
// wave owns 16 sample-rows x all 64 states. Dynamic LDS: 128 KB.
// ---------------------------------------------------------------------------
__global__ __launch_bounds__(256) void emission_kernel(
    const float* __restrict__ Y, const float* __restrict__ dT,
    const float* __restrict__ ws, float* __restrict__ ll) {
  extern __shared__ float sG[];   // [2][kD/2][kK][2] = 32768 floats

  const float* c0  = ws + OFF_C0;
  const float* am1 = ws + OFF_AM1;
  const float* bng = ws + OFF_BNEG;

  // ---- Stage the 128 KB weight block into LDS --------------------------
#if HAVE_TDM
  if (threadIdx.x < 32) {   // wave 0 issues one TDM op (EXEC is ignored by TDM)
    const unsigned lds_off = (unsigned)(size_t)(&sG[0]);
    const unsigned long long ga = (unsigned long long)(size_t)(ws + OFF_GP);
    // D# group 0: count=1 | lds_addr | global_addr[56:0] | type=2
    v4u g0;
    g0[0] = 1u;
    g0[1] = lds_off;
    g0[2] = (unsigned)ga;
    g0[3] = ((unsigned)(ga >> 32) & 0x1FFFFFFu) | 0x80000000u;
    // D# group 1: 1-D tensor, data_size=8B, dim0 = tile_dim0 = 16384 units
    const int dim0 = (int)(GP_SIZE / 2);   // 8-byte units
    v8i g1;
    g1[0] = (3 << 16);                      // data_size = 8B, mask=0
    g1[1] = (dim0 & 0xFFFF) << 16;          // tensor_dim0[15:0]
    g1[2] = (dim0 >> 16) | (1 << 16);       // tensor_dim0[31:16] | tensor_dim1=1
    g1[3] = (dim0 & 0xFFFF) << 16;          // tile_dim0
    g1[4] = 0;                              // tile_dim1/2 unused (1-D)
    g1[5] = dim0;                           // tensor_dim0_stride
    g1[6] = 0;
    g1[7] = 0;
    v4i g2 = {0, 0, 0, 0};
    v4i g3 = {0, 0, 0, 0};
    v8i g4 = {0, 0, 0, 0, 0, 0, 0, 0};
    __builtin_amdgcn_tensor_load_to_lds(g0, g1, g2, g3, g4, 0);
    __builtin_amdgcn_s_wait_tensorcnt(0);
  }
  __syncthreads();
#else
  {
    const float4* src = (const float4*)(ws + OFF_GP);
    float4* dst = (float4*)sG;
    for (int i = threadIdx.x; i < (int)(GP_SIZE / 4); i += 256) dst[i] = src[i];
  }
  __syncthreads();
#endif

  const int tid  = threadIdx.x;
  const int wave = tid >> 5;
  const int lane = tid & 31;
  const int half = lane >> 4;      // 0: K-slots {0,1}  1: K-slots {2,3}
  const int l15  = lane & 15;
  const int base = blockIdx.x * 128 + wave * 16;
  const int arow = base + l15;
  const int koff = half * 2;

  const float* Xr = Y + (size_t)arow * kD;

#if HAVE_WMMA_F32X4
  v8f acc[4] = {};

  for (int pass = 0; pass < 2; ++pass) {
    const float* Gs = sG + (size_t)pass * kD * kK;
    for (int p = 0; p < kD; p += 4) {
      float x0 = Xr[p + koff];
      float x1 = Xr[p + koff + 1];
      if (pass) { x0 *= x0; x1 *= x1; }
      v2f av; av.x = x0; av.y = x1;

      __builtin_prefetch(Xr + p + 32, 0, 3);

      // Paired layout: rows (p+koff, p+koff+1) x 16 states -> one b64 per tile
      const float* gp = Gs + (size_t)((p + koff) >> 1) * (2 * kK) + l15 * 2;
#pragma unroll
      for (int t = 0; t < 4; ++t) {
        v2f bv = *(const v2f*)(gp + t * 32);
        acc[t] = __builtin_amdgcn_wmma_f32_16x16x4_f32(
            false, av, false, bv, (short)0, acc[t], false, false);
      }
    }
  }

  // Epilogue: C layout -> row = base + v + half*8, col = t*16 + l15
  float dtv[8], ldv[8];
#pragma unroll
  for (int v = 0; v < 8; ++v) {
    const int r = base + half * 8 + v;
    const int b = r >> 13;
    const int n = r & (kN - 1);
    const float dt = dT[b * kN + n];
    dtv[v] = dt;
    ldv[v] = __logf(dt);
  }
#pragma unroll
  for (int t = 0; t < 4; ++t) {
    const int col = t * 16 + l15;
    const float c = c0[col], a1 = am1[col], bn = bng[col];
#pragma unroll
    for (int v = 0; v < 8; ++v) {
      const int r = base + half * 8 + v;
      ll[(size_t)r * kK + col] = acc[t][v] + c + a1 * ldv[v] + bn * dtv[v];
    }
  }
#else
  // Scalar fallback: each thread computes 32 (row,col) pairs from LDS weights.
  const int gt = blockIdx.x * 256 + tid;
  const int r  = gt >> 1;
  const int cb = (gt & 1) * 32;
  const int b  = r >> 13;
  const int n  = r & (kN - 1);
  const float dt = dT[b * kN + n];
  const float ld = __logf(dt);
  const float* Xr2 = Y + (size_t)r * kD;
  for (int col = cb; col < cb + 32; ++col) {
    float acc = 0.0f;
    for (int d = 0; d < kD; ++d) {
      const float y = Xr2[d];
      const int idx = (d >> 1) * (2 * kK) + col * 2 + (d & 1);
      acc = fmaf(y, sG[idx], acc);
      acc = fmaf(y * y, sG[kD * kK + idx], acc);
    }
    ll[(size_t)r * kK + col] = acc + c0[col] + am1[col] * ld + bng[col] * dt;
  }
#endif
}

// ---------------------------------------------------------------------------
// Kernel 3: forward scan. One 32-lane wave per batch, ZERO barriers in the
// hot loop: max via shfl_xor, e-broadcast via shfl, P pair-packed in LDS so
// each lane's two states come from one ds_load_b64.
// ---------------------------------------------------------------------------
__global__ __launch_bounds__(32) void scan_kernel(float* __restrict__ ws) {
  __shared__ float sP[kK * kK];   // 16 KB

  const int lane = threadIdx.x;
  const int b    = blockIdx.x;

  const float* PP  = ws + OFF_PP;
  const float* pi0 = ws + OFF_PI0;
  const float* llb = ws + OFF_LL + (size_t)b * kN * kK;

  for (int i = lane; i < kK * kK; i += 32) sP[i] = PP[i];
  __syncthreads();

  // lane owns states j0=lane and j1=lane+32
  float a0 = pi0[lane]      + llb[lane];
  float a1 = pi0[lane + 32] + llb[lane + 32];

  for (int t = 1; t < kN; ++t) {
    float m = fmaxf(a0, a1);
#pragma unroll
    for (int s = 16; s > 0; s >>= 1) m = fmaxf(m, __shfl_xor(m, s, 32));

    const float e0 = __expf(a0 - m);
    const float e1 = __expf(a1 - m);

    __builtin_prefetch(llb + (size_t)(t + 1) * kK + lane, 0, 3);

    float v0 = 0.0f, v1 = 0.0f;
#pragma unroll
    for (int i = 0; i < 32; ++i) {
      const float ei = __shfl(e0, i, 32);
      const v2f p = *(const v2f*)&sP[i * kK + lane * 2];
      v0 = fmaf(ei, p.x, v0);
      v1 = fmaf(ei, p.y, v1);
    }
#pragma unroll
    for (int i = 0; i < 32; ++i) {
      const float ei = __shfl(e1, i, 32);
      const v2f p = *(const v2f*)&sP[(i + 32) * kK + lane * 2];
      v0 = fmaf(ei, p.x, v0);
      v1 = fmaf(ei, p.y, v1);
    }

    a0 = m + __logf(v0) + llb[(size_t)t * kK + lane];
    a1 = m + __logf(v1) + llb[(size_t)t * kK + lane + 32];
  }

  // logZ_b = logsumexp over the 64 alphas
  float m = fmaxf(a0, a1);
#pragma unroll
  for (int s = 16; s > 0; s >>= 1) m = fmaxf(m, __shfl_xor(m, s, 32));
  float sum = __expf(a0 - m) + __expf(a1 - m);
#pragma unroll
  for (int s = 16; s > 0; s >>= 1) sum += __shfl_xor(sum, s, 32);
  if (lane == 0) ws[OFF_LOGZ + b] = m + __logf(sum);
}

// ---------------------------------------------------------------------------
// Kernel 4: loss = -sum_b logZ_b, appended after the Y copy in d_out
// ---------------------------------------------------------------------------
__global__ void finalize_kernel(const float* __restrict__ ws, float* __restrict__ out) {
  float s = 0.0f;
  for (int b = 0; b < kB; ++b) s += ws[OFF_LOGZ + b];
  out[(size_t)kB * kN * kD] = -s;
}

// ---------------------------------------------------------------------------
extern "C" void kernel_launch(void* const* d_in, const int* in_sizes, int n_in,
                              void* d_out, int out_size, void* d_ws, size_t ws_size,
                              hipStream_t stream) {
  const float* Y       = (const float*)d_in[0];
  const float* dT      = (const float*)d_in[1];
  const float* A       = (const float*)d_in[2];
  const float* log_ab  = (const float*)d_in[3];
  const float* mu      = (const float*)d_in[4];
  const float* log_sig = (const float*)d_in[5];
  const float* pi      = (const float*)d_in[6];
  float* out = (float*)d_out;
  float* ws  = (float*)d_ws;

  // Allow 128 KB dynamic LDS on the emission kernel (WGP has 320 KB).
  hipError_t attr_rc = hipFuncSetAttribute(
      reinterpret_cast<const void*>(emission_kernel),
      hipFuncAttributeMaxDynamicSharedMemorySize, (int)(GP_SIZE * sizeof(float)));
  (void)attr_rc;

  // Output 0 of the reference tuple is Y itself (identity embedding).
  hipError_t cp_rc = hipMemcpyAsync(d_out, d_in[0],
                                    sizeof(float) * (size_t)kB * kN * kD,
                                    hipMemcpyDeviceToDevice, stream);
  (void)cp_rc;

  prep_kernel<<<1, kK, 0, stream>>>(A, log_ab, mu, log_sig, pi, ws);
  emission_kernel<<<512, 256, (unsigned)(GP_SIZE * sizeof(float)), stream>>>(
      Y, dT, ws, ws + OFF_LL);
  scan_kernel<<<kB, 32, 0, stream>>>(ws);
  finalize_kernel<<<1, 1, 0, stream>>>(ws, out);
}